// WDNet_39728447488766
// MI455X (gfx1250) — compile-verified
//
#include <hip/hip_runtime.h>
#include <math.h>

#define BB   4
#define CC   64
#define HIDN 256
#define PP   64
#define HH   128
#define WWD  128
#define HWN  16384
#define EPS_ 1e-8f

typedef __attribute__((ext_vector_type(2))) float v2f;
typedef __attribute__((ext_vector_type(8))) float v8f;

__device__ __forceinline__ v8f v8zero() {
    v8f z;
    #pragma unroll
    for (int i = 0; i < 8; ++i) z[i] = 0.f;
    return z;
}

__device__ __forceinline__ v8f wmma4(v2f a, v2f b, v8f c) {
    // D = A(16x4,f32) * B(4x16,f32) + C(16x16,f32)
    return __builtin_amdgcn_wmma_f32_16x16x4_f32(false, a, false, b, (short)0, c, false, false);
}

// Async global->LDS copies (ASYNCcnt-tracked DMA, CDNA5 §15.18.3 op 96/98).
// LDS operand is the 32-bit LDS byte offset (= low 32 bits of generic addr).
__device__ __forceinline__ void async_ld_b32(void* lds, const void* g) {
    asm volatile("global_load_async_to_lds_b32 %0, %1, off"
                 :: "v"((unsigned)(unsigned long long)lds), "v"(g) : "memory");
}
__device__ __forceinline__ void async_ld_b128(void* lds, const void* g) {
    asm volatile("global_load_async_to_lds_b128 %0, %1, off"
                 :: "v"((unsigned)(unsigned long long)lds), "v"(g) : "memory");
}
__device__ __forceinline__ void wait_async0() {
    asm volatile("s_wait_asynccnt 0x0" ::: "memory");
}

// ---------------------------------------------------------------------------
// Kernel 1: t1[b,hid,p] = sum_c x[b,c,p] * W1[c,hid] + b1[hid]
// Block tile: 64 (hid) x 64 (px), 4 waves, K = 64. X-tile via async DMA.
// ---------------------------------------------------------------------------
__global__ __launch_bounds__(128) void k_gemm1(const float* __restrict__ x,
                                               const float* __restrict__ W1,
                                               const float* __restrict__ b1,
                                               float* __restrict__ t1) {
    __shared__ float ldsX[64 * 64];
    const int b  = blockIdx.z;
    const int m0 = blockIdx.y * 64;
    const int p0 = blockIdx.x * 64;
    const int tid = threadIdx.x;

    for (int ch = tid; ch < 1024; ch += 128) {          // 16B chunks
        int k = ch >> 4, p4 = (ch & 15) << 2;
        async_ld_b128(&ldsX[k * 64 + p4],
                      &x[((size_t)(b * CC + k)) * HWN + p0 + p4]);
    }
    wait_async0();
    __syncthreads();

    const int lane = tid & 31, wave = tid >> 5;
    const int l15 = lane & 15, hi = lane >> 4;
    const int mt = wave;
    v8f acc[4];
    #pragma unroll
    for (int nt = 0; nt < 4; ++nt) acc[nt] = v8zero();

    const int m = m0 + mt * 16 + l15;
    for (int k0 = 0; k0 < 64; k0 += 4) {
        int kb = k0 + 2 * hi;
        v2f a;
        a.x = W1[kb * HIDN + m];
        a.y = W1[(kb + 1) * HIDN + m];
        #pragma unroll
        for (int nt = 0; nt < 4; ++nt) {
            v2f bf;
            bf.x = ldsX[kb * 64 + nt * 16 + l15];
            bf.y = ldsX[(kb + 1) * 64 + nt * 16 + l15];
            acc[nt] = wmma4(a, bf, acc[nt]);
        }
    }
    #pragma unroll
    for (int nt = 0; nt < 4; ++nt) {
        #pragma unroll
        for (int g = 0; g < 8; ++g) {
            int hid = m0 + mt * 16 + g + 8 * hi;
            int col = p0 + nt * 16 + l15;
            t1[((size_t)(b * HIDN + hid)) * HWN + col] = acc[nt][g] + b1[hid];
        }
    }
}

// ---------------------------------------------------------------------------
// Kernel 2: per-(b,ch) min/max over HW.
// ---------------------------------------------------------------------------
__global__ __launch_bounds__(256) void k_minmax(const float* __restrict__ t1,
                                                float* __restrict__ mm) {
    __shared__ float smin[256], smax[256];
    const int bc = blockIdx.x;
    const float* src = t1 + (size_t)bc * HWN;
    float mn = 3.4e38f, mx = -3.4e38f;
    for (int i = threadIdx.x; i < HWN; i += 256) {
        float v = src[i];
        mn = fminf(mn, v);
        mx = fmaxf(mx, v);
    }
    smin[threadIdx.x] = mn;
    smax[threadIdx.x] = mx;
    __syncthreads();
    for (int s = 128; s > 0; s >>= 1) {
        if (threadIdx.x < s) {
            smin[threadIdx.x] = fminf(smin[threadIdx.x], smin[threadIdx.x + s]);
            smax[threadIdx.x] = fmaxf(smax[threadIdx.x], smax[threadIdx.x + s]);
        }
        __syncthreads();
    }
    if (threadIdx.x == 0) { mm[2 * bc] = smin[0]; mm[2 * bc + 1] = smax[0]; }
}

// ---------------------------------------------------------------------------
// Kernel 3: 256-bin histogram + entropy activity per (b,ch).
// ---------------------------------------------------------------------------
__global__ __launch_bounds__(256) void k_hist(const float* __restrict__ t1,
                                              const float* __restrict__ mm,
                                              float* __restrict__ activity) {
    __shared__ unsigned h[256];
    __shared__ float red[256];
    const int bc = blockIdx.x;
    h[threadIdx.x] = 0u;
    __syncthreads();
    const float mn = mm[2 * bc], mx = mm[2 * bc + 1];
    const float inv = 1.f / (mx - mn + EPS_);
    const float* src = t1 + (size_t)bc * HWN;
    for (int i = threadIdx.x; i < HWN; i += 256) {
        float xn = (src[i] - mn) * inv;
        int iv = (int)(xn * 256.f);          // trunc toward zero (xn >= 0)
        iv = iv < 0 ? 0 : (iv > 255 ? 255 : iv);
        atomicAdd(&h[iv], 1u);
    }
    __syncthreads();
    const float S = (float)HWN + 256.f * EPS_;
    float prob = ((float)h[threadIdx.x] + EPS_) / S;
    red[threadIdx.x] = -prob * logf(prob + EPS_);
    __syncthreads();
    for (int s = 128; s > 0; s >>= 1) {
        if (threadIdx.x < s) red[threadIdx.x] += red[threadIdx.x + s];
        __syncthreads();
    }
    if (threadIdx.x == 0) activity[bc] = red[0];
}

// ---------------------------------------------------------------------------
// Kernel 4: top-64 by activity (descending, lowest-index tie-break like
// jax.lax.top_k) + ascending unselected list (matches argsort construction).
// ---------------------------------------------------------------------------
__global__ void k_select(const float* __restrict__ activity,
                         int* __restrict__ sel, int* __restrict__ unsel) {
    const int b = blockIdx.x;
    if (threadIdx.x != 0) return;
    const float* act = activity + b * HIDN;
    bool used[HIDN];
    for (int i = 0; i < HIDN; ++i) used[i] = false;
    for (int j = 0; j < PP; ++j) {
        float best = -3.4e38f; int bi = 0;
        for (int i = 0; i < HIDN; ++i)
            if (!used[i] && act[i] > best) { best = act[i]; bi = i; }
        used[bi] = true;
        sel[b * PP + j] = bi;
    }
    int u = 0;
    for (int i = 0; i < HIDN; ++i)
        if (!used[i]) unsel[b * (HIDN - PP) + (u++)] = i;
}

// ---------------------------------------------------------------------------
// Kernel 5: 3x3 conv on 64 selected channels as 9-tap WMMA GEMM.
// Block = (b, row y, 64-px x segment); M=64 out-ch, N=64 px, K=64 in-ch.
// LDS tile zero-filled, then in-bounds elements DMA'd (EXEC-masked async).
// ---------------------------------------------------------------------------
__global__ __launch_bounds__(128) void k_conv(const float* __restrict__ t1,
                                              const float* __restrict__ cw,
                                              const float* __restrict__ cb,
                                              const int* __restrict__ sel,
                                              float* __restrict__ part) {
    __shared__ float ldsIn[64 * 3 * 68];   // [ch][row 0..2][x halo 0..65]
    __shared__ int ldsSel[64];
    const int b = blockIdx.z, y = blockIdx.y, x0 = blockIdx.x * 64;
    const int tid = threadIdx.x;

    if (tid < 64) ldsSel[tid] = sel[b * PP + tid];
    for (int idx = tid; idx < 64 * 3 * 68; idx += 128) ldsIn[idx] = 0.f;
    __syncthreads();

    for (int idx = tid; idx < 64 * 3 * 66; idx += 128) {
        int k  = idx / 198;
        int r2 = idx - k * 198;
        int r  = r2 / 66;
        int xi = r2 - r * 66;
        int gy = y + r - 1;
        int gx = x0 - 1 + xi;
        if (gy >= 0 && gy < HH && gx >= 0 && gx < WWD)
            async_ld_b32(&ldsIn[(k * 3 + r) * 68 + xi],
                         &t1[((size_t)(b * HIDN + ldsSel[k])) * HWN + gy * WWD + gx]);
    }
    wait_async0();
    __syncthreads();

    const int lane = tid & 31, wave = tid >> 5;
    const int l15 = lane & 15, hi = lane >> 4;
    const int mt = wave;
    v8f acc[4];
    #pragma unroll
    for (int nt = 0; nt < 4; ++nt) acc[nt] = v8zero();

    const int m = mt * 16 + l15;
    for (int ky = 0; ky < 3; ++ky) {
        for (int kx = 0; kx < 3; ++kx) {
            for (int k0 = 0; k0 < 64; k0 += 4) {
                int kb = k0 + 2 * hi;
                v2f a;
                a.x = cw[m * 576 + kb * 9 + ky * 3 + kx];
                a.y = cw[m * 576 + (kb + 1) * 9 + ky * 3 + kx];
                #pragma unroll
                for (int nt = 0; nt < 4; ++nt) {
                    int c = nt * 16 + l15;
                    v2f bf;
                    bf.x = ldsIn[(kb * 3 + ky) * 68 + c + kx];
                    bf.y = ldsIn[((kb + 1) * 3 + ky) * 68 + c + kx];
                    acc[nt] = wmma4(a, bf, acc[nt]);
                }
            }
        }
    }
    #pragma unroll
    for (int nt = 0; nt < 4; ++nt) {
        #pragma unroll
        for (int g = 0; g < 8; ++g) {
            int o  = mt * 16 + g + 8 * hi;
            int xc = x0 + nt * 16 + l15;
            part[((size_t)(b * PP + o)) * HWN + y * WWD + xc] = acc[nt][g] + cb[o];
        }
    }
}

// ---------------------------------------------------------------------------
// Kernel 6: gather -> (relu at A-frag read) -> GEMM2 (M=32 px, N=64 ch,
// K=256) -> bias, relu, residual, LayerNorm, gamma/beta, NCHW store.
// Gather staged via async DMA; LDS A-tile reused for LN stage.
// ---------------------------------------------------------------------------
__global__ __launch_bounds__(128) void k_final(const float* __restrict__ x,
                                               const float* __restrict__ W2,
                                               const float* __restrict__ b2,
                                               const float* __restrict__ t1,
                                               const float* __restrict__ part,
                                               const int* __restrict__ unsel,
                                               const float* __restrict__ gamma,
                                               const float* __restrict__ beta,
                                               float* __restrict__ out) {
    __shared__ float smem[HIDN * 32];      // 32 KB: A-tile, then LN buffer
    __shared__ int ldsMap[HIDN - PP];
    const int b = blockIdx.y;
    const int p0 = blockIdx.x * 32;
    const int tid = threadIdx.x;

    for (int i = tid; i < HIDN - PP; i += 128) ldsMap[i] = unsel[b * (HIDN - PP) + i];
    __syncthreads();

    for (int ch = tid; ch < HIDN * 8; ch += 128) {      // 16B chunks
        int k = ch >> 3, p4 = (ch & 7) << 2;
        const float* src = (k < PP)
            ? &part[((size_t)(b * PP + k)) * HWN + p0 + p4]
            : &t1[((size_t)(b * HIDN + ldsMap[k - PP])) * HWN + p0 + p4];
        async_ld_b128(&smem[k * 32 + p4], src);
    }
    wait_async0();
    __syncthreads();

    const int lane = tid & 31, wave = tid >> 5;
    const int l15 = lane & 15, hi = lane >> 4;
    const int pt = wave & 1, ntb = (wave >> 1) * 2;
    v8f acc[2];
    acc[0] = v8zero(); acc[1] = v8zero();

    for (int k0 = 0; k0 < HIDN; k0 += 4) {
        int kb = k0 + 2 * hi;
        v2f a;                                            // relu folded in
        a.x = fmaxf(smem[kb * 32 + pt * 16 + l15], 0.f);
        a.y = fmaxf(smem[(kb + 1) * 32 + pt * 16 + l15], 0.f);
        #pragma unroll
        for (int t = 0; t < 2; ++t) {
            int c = (ntb + t) * 16 + l15;
            v2f bf;
            bf.x = W2[kb * CC + c];
            bf.y = W2[(kb + 1) * CC + c];
            acc[t] = wmma4(a, bf, acc[t]);
        }
    }
    __syncthreads();   // all A-tile reads done; reuse smem for LN staging

    #pragma unroll
    for (int t = 0; t < 2; ++t) {
        #pragma unroll
        for (int g = 0; g < 8; ++g) {
            int lp   = pt * 16 + g + 8 * hi;
            int chan = (ntb + t) * 16 + l15;
            float u = fmaxf(acc[t][g] + b2[chan], 0.f)
                    + x[((size_t)(b * CC + chan)) * HWN + p0 + lp];
            smem[lp * 65 + chan] = u;
        }
    }
    __syncthreads();

    if (tid < 32) {
        const float* row = smem + tid * 65;
        float mu = 0.f;
        for (int c = 0; c < CC; ++c) mu += row[c];
        mu *= (1.f / CC);
        float var = 0.f;
        for (int c = 0; c < CC; ++c) { float d = row[c] - mu; var += d * d; }
        var *= (1.f / CC);
        float inv = rsqrtf(var + 1e-5f);
        for (int c = 0; c < CC; ++c)
            out[((size_t)(b * CC + c)) * HWN + p0 + tid] =
                (row[c] - mu) * inv * gamma[c] + beta[c];
    }
}

// ---------------------------------------------------------------------------
extern "C" void kernel_launch(void* const* d_in, const int* in_sizes, int n_in,
                              void* d_out, int out_size, void* d_ws, size_t ws_size,
                              hipStream_t stream) {
    const float* x     = (const float*)d_in[0];
    const float* W1    = (const float*)d_in[1];
    const float* b1    = (const float*)d_in[2];
    const float* W2    = (const float*)d_in[3];
    const float* b2    = (const float*)d_in[4];
    const float* cw    = (const float*)d_in[5];
    const float* cb    = (const float*)d_in[6];
    const float* gamma = (const float*)d_in[7];
    const float* beta  = (const float*)d_in[8];
    float* out = (float*)d_out;

    float* t1       = (float*)d_ws;                      // B*HID*HW
    float* part     = t1 + (size_t)BB * HIDN * HWN;      // B*P*HW
    float* mm       = part + (size_t)BB * PP * HWN;      // 2*B*HID
    float* activity = mm + 2 * BB * HIDN;                // B*HID
    int*   sel      = (int*)(activity + BB * HIDN);      // B*P
    int*   unsel    = sel + BB * PP;                     // B*(HID-P)

    k_gemm1 <<<dim3(HWN / 64, HIDN / 64, BB), 128, 0, stream>>>(x, W1, b1, t1);
    k_minmax<<<BB * HIDN, 256, 0, stream>>>(t1, mm);
    k_hist  <<<BB * HIDN, 256, 0, stream>>>(t1, mm, activity);
    k_select<<<BB, 32, 0, stream>>>(activity, sel, unsel);
    k_conv  <<<dim3(WWD / 64, HH, BB), 128, 0, stream>>>(t1, cw, cb, sel, part);
    k_final <<<dim3(HWN / 32, BB), 128, 0, stream>>>(x, W2, b2, t1, part, unsel,
                                                     gamma, beta, out);
}